// GIN_4303557230930
// MI455X (gfx1250) — compile-verified
//
#include <hip/hip_runtime.h>

#define NN 100000
#define EE 800000
#define DD 128
#define LL 3

typedef float v2f __attribute__((ext_vector_type(2)));
typedef float v8f __attribute__((ext_vector_type(8)));

// ---------------------------------------------------------------- prep: agg = X, stats = 0
__global__ void gin_prep_kernel(const float* __restrict__ X, float* __restrict__ agg,
                                float* __restrict__ stats) {
    size_t idx = (size_t)blockIdx.x * blockDim.x + threadIdx.x;
    const float4* s = (const float4*)X;
    float4* d = (float4*)agg;
    d[idx] = s[idx];
    if (blockIdx.x == 0 && threadIdx.x < 2 * DD) stats[threadIdx.x] = 0.0f;
}

// ---------------------------------------------------------------- repack W into WMMA-B fragment order
// Fragment f = ((j*32 + kk)*32 + lane): lanes 0-15 -> (K=4kk+0..1, N=16j+lane),
// lanes 16-31 -> (K=4kk+2..3, N=16j+lane-16). Stored as float2 so the MLP kernel
// fetches each B fragment with one coalesced global_load_b64.
__global__ void gin_pack_kernel(const float* __restrict__ W1, const float* __restrict__ W2,
                                v2f* __restrict__ Wp) {
    int f = blockIdx.x * 256 + threadIdx.x;   // [0, 6*8192)
    int m = f >> 13;                          // matrix 0..5 (3x W1, 3x W2)
    int t = f & 8191;
    int j    = t >> 10;
    int kk   = (t >> 5) & 31;
    int lane = t & 31;
    int koff = (lane >> 4) << 1;
    int n    = 16 * j + (lane & 15);
    int krow = 4 * kk + koff;
    const float* W = (m < 3) ? (W1 + m * DD * DD) : (W2 + (m - 3) * DD * DD);
    v2f v;
    v.x = W[krow * DD + n];
    v.y = W[(krow + 1) * DD + n];
    Wp[f] = v;
}

// ---------------------------------------------------------------- scatter-add over edges
__global__ void gin_scatter_kernel(const float* __restrict__ X, const int* __restrict__ ei,
                                   float* __restrict__ agg) {
    unsigned g = blockIdx.x * 256u + threadIdx.x;
    unsigned e = g >> 5;           // edge id (32 lanes per edge)
    unsigned c = (g & 31u) << 2;   // channel quad
    int s = ei[e];
    int d = ei[EE + e];
    const float4 v = *(const float4*)(X + (size_t)s * DD + c);
    float* p = agg + (size_t)d * DD + c;
    atomicAdd(p + 0, v.x);
    atomicAdd(p + 1, v.y);
    atomicAdd(p + 2, v.z);
    atomicAdd(p + 3, v.w);
}

// ---------------------------------------------------------------- fused MLP: Z = relu(A@W1+b1)@W2+b2
__global__ __launch_bounds__(256)
void gin_mlp_kernel(const float* __restrict__ A,
                    const v2f* __restrict__ Wp1, const float* __restrict__ b1,
                    const v2f* __restrict__ Wp2, const float* __restrict__ b2,
                    float* __restrict__ Z) {
    __shared__ float ts[8][16][DD];          // 64 KB: per-wave 16x128 staging of t
    const int w    = threadIdx.x >> 5;
    const int lane = threadIdx.x & 31;
    const int NT   = NN / 16;                // 6250 tiles
    int tile = blockIdx.x * 8 + w;
    if (tile >= NT) tile = NT - 1;           // tail waves duplicate last tile (identical writes)
    const int rowBase = tile * 16;

    const int r    = lane & 15;              // A: row / B,C: column-in-tile
    const int koff = (lane >> 4) << 1;       // lanes 16-31 hold K+2,K+3
    const int mh   = (lane >> 4) << 3;       // C: lanes 16-31 hold M+8

    __builtin_prefetch(Wp1, 0, 1);           // warm packed weights (global_prefetch_b8)
    __builtin_prefetch(Wp2, 0, 1);

    v2f a[32];

    // ---- GEMM1: t = relu(A @ W1 + b1) -> LDS ----
    {
        const float* Ap = A + (size_t)(rowBase + r) * DD + koff;
        #pragma unroll
        for (int kk = 0; kk < 32; ++kk)
            a[kk] = *(const v2f*)(Ap + 4 * kk);
    }
    #pragma unroll 1
    for (int j = 0; j < 8; j += 2) {
        const v2f* B0 = Wp1 + (j * 32) * 32 + lane;        // fragment stream, tile j
        const v2f* B1 = Wp1 + ((j + 1) * 32) * 32 + lane;  // fragment stream, tile j+1
        v8f c0 = {}, c1 = {};
        #pragma unroll
        for (int kk = 0; kk < 32; ++kk) {
            v2f bf0 = B0[kk * 32];
            v2f bf1 = B1[kk * 32];
            c0 = __builtin_amdgcn_wmma_f32_16x16x4_f32(false, a[kk], false, bf0,
                                                       (short)0, c0, false, false);
            c1 = __builtin_amdgcn_wmma_f32_16x16x4_f32(false, a[kk], false, bf1,
                                                       (short)0, c1, false, false);
        }
        const float bias0 = b1[16 * j + r];
        const float bias1 = b1[16 * (j + 1) + r];
        #pragma unroll
        for (int rv = 0; rv < 8; ++rv) {
            ts[w][mh + rv][16 * j + r]       = fmaxf(c0[rv] + bias0, 0.0f);
            ts[w][mh + rv][16 * (j + 1) + r] = fmaxf(c1[rv] + bias1, 0.0f);
        }
    }

    // ---- GEMM2: z = t @ W2 + b2 -> global ----
    {
        #pragma unroll
        for (int kk = 0; kk < 32; ++kk)
            a[kk] = *(const v2f*)&ts[w][r][4 * kk + koff];
    }
    #pragma unroll 1
    for (int j = 0; j < 8; j += 2) {
        const v2f* B0 = Wp2 + (j * 32) * 32 + lane;
        const v2f* B1 = Wp2 + ((j + 1) * 32) * 32 + lane;
        v8f c0 = {}, c1 = {};
        #pragma unroll
        for (int kk = 0; kk < 32; ++kk) {
            v2f bf0 = B0[kk * 32];
            v2f bf1 = B1[kk * 32];
            c0 = __builtin_amdgcn_wmma_f32_16x16x4_f32(false, a[kk], false, bf0,
                                                       (short)0, c0, false, false);
            c1 = __builtin_amdgcn_wmma_f32_16x16x4_f32(false, a[kk], false, bf1,
                                                       (short)0, c1, false, false);
        }
        const float bias0 = b2[16 * j + r];
        const float bias1 = b2[16 * (j + 1) + r];
        #pragma unroll
        for (int rv = 0; rv < 8; ++rv) {
            Z[(size_t)(rowBase + mh + rv) * DD + 16 * j + r]       = c0[rv] + bias0;
            Z[(size_t)(rowBase + mh + rv) * DD + 16 * (j + 1) + r] = c1[rv] + bias1;
        }
    }
}

// ---------------------------------------------------------------- per-channel sum / sumsq
__global__ void gin_stats_kernel(const float* __restrict__ Z, float* __restrict__ stats) {
    const int c  = threadIdx.x & (DD - 1);
    const int ro = threadIdx.x >> 7;     // 0 or 1
    const int base = blockIdx.x * 256;
    float s = 0.0f, q = 0.0f;
    for (int rr = base + ro; rr < base + 256 && rr < NN; rr += 2) {
        float v = Z[(size_t)rr * DD + c];
        s += v;
        q += v * v;
    }
    atomicAdd(&stats[c], s);
    atomicAdd(&stats[DD + c], q);
}

// ---------------------------------------------------------------- batchnorm (biased var) + relu
__global__ void gin_bn_kernel(const float* __restrict__ Z, const float* __restrict__ stats,
                              const float* __restrict__ gamma, const float* __restrict__ beta,
                              float* __restrict__ Out) {
    size_t idx = (size_t)blockIdx.x * 256 + threadIdx.x;
    int c = (int)(idx & (DD - 1));
    float mean = stats[c] * (1.0f / NN);
    float var  = stats[DD + c] * (1.0f / NN) - mean * mean;
    float inv  = rsqrtf(var + 1e-5f);
    float v    = gamma[c] * (Z[idx] - mean) * inv + beta[c];
    Out[idx]   = fmaxf(v, 0.0f);
}

extern "C" void kernel_launch(void* const* d_in, const int* in_sizes, int n_in,
                              void* d_out, int out_size, void* d_ws, size_t ws_size,
                              hipStream_t stream) {
    const float* x     = (const float*)d_in[0];
    const int*   ei    = (const int*)d_in[1];
    const float* W1    = (const float*)d_in[2];
    const float* b1    = (const float*)d_in[3];
    const float* W2    = (const float*)d_in[4];
    const float* b2    = (const float*)d_in[5];
    const float* gamma = (const float*)d_in[6];
    const float* beta  = (const float*)d_in[7];

    float* zbuf  = (float*)d_out;                       // z lives in d_out
    float* hbuf  = (float*)d_ws;                        // hidden state
    float* agg   = hbuf + (size_t)NN * DD;              // aggregation buffer
    float* stats = agg + (size_t)NN * DD;               // 256 floats
    v2f*   wpack = (v2f*)(stats + 256);                 // 6 x 8192 packed fragments

    const int PREP_blocks = (NN * DD / 4) / 256;        // 12500
    const int PACK_blocks = (6 * 8192) / 256;           // 192
    const int SC_blocks   = (EE * 32) / 256;            // 100000
    const int MLP_blocks  = ((NN / 16) + 7) / 8;        // 782
    const int ST_blocks   = (NN + 255) / 256;           // 391
    const int BN_blocks   = (NN * DD) / 256;            // 50000

    gin_pack_kernel<<<PACK_blocks, 256, 0, stream>>>(W1, W2, wpack);

    for (int i = 0; i < LL; ++i) {
        const float* X = (i == 0) ? x : hbuf;
        gin_prep_kernel<<<PREP_blocks, 256, 0, stream>>>(X, agg, stats);
        gin_scatter_kernel<<<SC_blocks, 256, 0, stream>>>(X, ei, agg);
        gin_mlp_kernel<<<MLP_blocks, 256, 0, stream>>>(agg, wpack + i * 8192, b1 + i * DD,
                                                       wpack + (3 + i) * 8192, b2 + i * DD, zbuf);
        gin_stats_kernel<<<ST_blocks, 256, 0, stream>>>(zbuf, stats);
        gin_bn_kernel<<<BN_blocks, 256, 0, stream>>>(zbuf, stats, gamma + i * DD, beta + i * DD,
                                                     (i < LL - 1) ? hbuf : zbuf);
    }
    (void)in_sizes; (void)n_in; (void)out_size; (void)ws_size;
}